// HybridFocalLoss_75986561401207
// MI455X (gfx1250) — compile-verified
//
#include <hip/hip_runtime.h>
#include <hip/hip_bf16.h>
#include <stdint.h>

// ---- problem constants (match reference) ----
#define B_ROWS   32768
#define C_COLS   1000
#define ROW_F4   250          // 1000 floats = 250 float4 per row
#define WPB      8            // waves per block (wave32)
#define BLOCK_T  256
#define GRID_B   (B_ROWS / WPB)   // 4096 blocks -> exactly one wave per row
#define LOG2E_F  1.4426950408889634f
#define LN2_F    0.6931471805599453f

// ---------------- CDNA5 async global->LDS copy ----------------
// clang prototype (probe-confirmed via diagnostic):
//   void __builtin_amdgcn_global_load_async_to_lds_b128(
//       v4i addrspace(1)* gsrc, v4i addrspace(3)* lds_dst, imm offset, imm cpol)
typedef int v4i_t __attribute__((vector_size(16)));
typedef __attribute__((address_space(1))) v4i_t* gas_b128_t;
typedef __attribute__((address_space(3))) v4i_t* las_b128_t;

__device__ __forceinline__ void async_copy_b128(void* lds_dst, const void* gsrc) {
#if __has_builtin(__builtin_amdgcn_global_load_async_to_lds_b128)
  __builtin_amdgcn_global_load_async_to_lds_b128(
      (gas_b128_t)(gsrc), (las_b128_t)(lds_dst), /*offset=*/0, /*cpol=*/0);
#else
  unsigned lds_off = (unsigned)(uintptr_t)((las_b128_t)(lds_dst));
  unsigned long long ga = (unsigned long long)(uintptr_t)gsrc;
  asm volatile("global_load_async_to_lds_b128 %0, %1, off"
               :: "v"(lds_off), "v"(ga) : "memory");
#endif
}

__device__ __forceinline__ void wait_async_zero() {
  asm volatile("s_wait_asynccnt 0" ::: "memory");
}

// ---------------- wave32 reductions ----------------
__device__ __forceinline__ float wave_sum(float v) {
#pragma unroll
  for (int m = 16; m > 0; m >>= 1) v += __shfl_xor(v, m, 32);
  return v;
}
__device__ __forceinline__ float wave_max(float v) {
#pragma unroll
  for (int m = 16; m > 0; m >>= 1) v = fmaxf(v, __shfl_xor(v, m, 32));
  return v;
}

// ---------------- kernel: one wave per row (defined FIRST so disasm shows it) ----
__global__ __launch_bounds__(BLOCK_T)
void hfl_main(const float* __restrict__ inp, const int* __restrict__ tgt,
              float* __restrict__ cls_sum, float* __restrict__ cls_cnt,
              float* __restrict__ part_focal, float* __restrict__ part_sq) {
  __shared__ float4 lbuf[WPB][ROW_F4];     // 32000 B
  __shared__ float  sfoc[WPB];
  __shared__ float  ssq[WPB];

  const int lane = threadIdx.x & 31;
  const int wid  = threadIdx.x >> 5;
  const int row  = blockIdx.x * WPB + wid;           // always < B_ROWS

  const float4* grow = (const float4*)inp + (size_t)row * ROW_F4;
  float4* lrow = &lbuf[wid][0];

  // ---- stage one row into LDS via async DMA (8 x b128 per lane max) ----
#pragma unroll
  for (int k = 0; k < 8; ++k) {
    int idx = lane + 32 * k;
    if (idx < ROW_F4) async_copy_b128(lrow + idx, grow + idx);
  }
  wait_async_zero();

  // ---- pass 1 out of LDS: max, sum(x), sum(x^2) ----
  float mx = -3.0e38f, sx = 0.0f, sq = 0.0f;
#pragma unroll
  for (int k = 0; k < 8; ++k) {
    int idx = lane + 32 * k;
    if (idx < ROW_F4) {
      float4 v = lrow[idx];
      sx += v.x + v.y + v.z + v.w;
      sq = fmaf(v.x, v.x, sq); sq = fmaf(v.y, v.y, sq);
      sq = fmaf(v.z, v.z, sq); sq = fmaf(v.w, v.w, sq);
      mx = fmaxf(mx, fmaxf(fmaxf(v.x, v.y), fmaxf(v.z, v.w)));
    }
  }
  mx = wave_max(mx);
  sx = wave_sum(sx);
  sq = wave_sum(sq);

  // ---- pass 2: sum exp(x - mx) in base-2 domain ----
  const float m2y = mx * LOG2E_F;
  float s = 0.0f;
#pragma unroll
  for (int k = 0; k < 8; ++k) {
    int idx = lane + 32 * k;
    if (idx < ROW_F4) {
      float4 v = lrow[idx];
      s += __builtin_amdgcn_exp2f(fmaf(v.x, LOG2E_F, -m2y));
      s += __builtin_amdgcn_exp2f(fmaf(v.y, LOG2E_F, -m2y));
      s += __builtin_amdgcn_exp2f(fmaf(v.z, LOG2E_F, -m2y));
      s += __builtin_amdgcn_exp2f(fmaf(v.w, LOG2E_F, -m2y));
    }
  }
  s = wave_sum(s);

  if (lane == 0) {
    const int t = tgt[row];
    const float xt  = ((const float*)lrow)[t];       // gathered logit (from LDS)
    const float lse = fmaf(__builtin_amdgcn_logf(s), LN2_F, mx);  // mx + ln(sumexp)
    const float nll    = lse - xt;
    const float smooth = lse - sx * (1.0f / (float)C_COLS);
    const float ce = 0.9f * nll + 0.1f * smooth;     // label smoothing 0.1
    const float pt = __builtin_amdgcn_exp2f(-ce * LOG2E_F);
    const float om = 1.0f - pt;
    const float focal = om * om * ce;                // ALPHA=1, GAMMA=2

    atomicAdd(&cls_sum[t], xt);                      // segment sums (1000 classes)
    atomicAdd(&cls_cnt[t], 1.0f);                    // exact: small integers in fp32
    sfoc[wid] = focal;
    ssq[wid]  = sq;
  }
  __syncthreads();

  if (threadIdx.x == 0) {                            // deterministic per-block partials
    float f = 0.0f, q = 0.0f;
#pragma unroll
    for (int i = 0; i < WPB; ++i) { f += sfoc[i]; q += ssq[i]; }
    part_focal[blockIdx.x] = f;
    part_sq[blockIdx.x]    = q;
  }
}

// ---------------- kernel: zero per-class accumulators ----------------
__global__ void hfl_zero(float* p, int n) {
  int i = blockIdx.x * blockDim.x + threadIdx.x;
  if (i < n) p[i] = 0.0f;
}

// ---------------- kernel: finalize ----------------
__global__ __launch_bounds__(1024)
void hfl_finalize(const float* __restrict__ cls_sum, const float* __restrict__ cls_cnt,
                  const float* __restrict__ part_focal, const float* __restrict__ part_sq,
                  float* __restrict__ out) {
  const int tid = threadIdx.x;

  float cross = 0.0f, msq = 0.0f;
  if (tid < C_COLS) {
    float sc = cls_sum[tid];
    float nc = cls_cnt[tid];
    float m  = (nc > 0.0f) ? sc / nc : 0.0f;
    cross = m * sc;          // sum_i gathered_i * m_{t_i}
    msq   = nc * m * m;      // sum_i m_{t_i}^2
  }
  float f = 0.0f, q = 0.0f;
  for (int i = tid; i < GRID_B; i += 1024) { f += part_focal[i]; q += part_sq[i]; }

  cross = wave_sum(cross); msq = wave_sum(msq);
  f = wave_sum(f);         q = wave_sum(q);

  __shared__ float red[4][32];
  const int lane = tid & 31, wid = tid >> 5;
  if (lane == 0) { red[0][wid] = cross; red[1][wid] = msq; red[2][wid] = f; red[3][wid] = q; }
  __syncthreads();

  if (tid == 0) {
    float c = 0.0f, mm = 0.0f, ff = 0.0f, qq = 0.0f;
#pragma unroll
    for (int i = 0; i < 32; ++i) { c += red[0][i]; mm += red[1][i]; ff += red[2][i]; qq += red[3][i]; }
    const float inv_n = 1.0f / ((float)B_ROWS * (float)C_COLS);
    const float center = (qq - 2.0f * c + mm) * inv_n;
    out[0] = ff * (1.0f / (float)B_ROWS) + 0.1f * center;  // CENTER_W = 0.1
  }
}

// ---------------- launcher ----------------
extern "C" void kernel_launch(void* const* d_in, const int* in_sizes, int n_in,
                              void* d_out, int out_size, void* d_ws, size_t ws_size,
                              hipStream_t stream) {
  (void)in_sizes; (void)n_in; (void)out_size; (void)ws_size;
  const float* inp = (const float*)d_in[0];
  const int*   tgt = (const int*)d_in[1];
  float* ws_f      = (float*)d_ws;

  float* cls_sum    = ws_f;               // [1024]
  float* cls_cnt    = ws_f + 1024;        // [1024]
  float* part_focal = ws_f + 2048;        // [4096]
  float* part_sq    = ws_f + 2048 + 4096; // [4096]
  float* out        = (float*)d_out;

  hfl_zero<<<8, 256, 0, stream>>>(ws_f, 2048);
  hfl_main<<<GRID_B, BLOCK_T, 0, stream>>>(inp, tgt, cls_sum, cls_cnt,
                                           part_focal, part_sq);
  hfl_finalize<<<1, 1024, 0, stream>>>(cls_sum, cls_cnt, part_focal, part_sq, out);
}